// VectorQuantizer_19456201851525
// MI455X (gfx1250) — compile-verified
//
#include <hip/hip_runtime.h>
#include <hip/hip_bf16.h>

typedef __attribute__((ext_vector_type(2))) float v2f;
typedef __attribute__((ext_vector_type(8))) float v8f;

#define NROWS   65536      // 16*64*64 vectors
#define DIM     64
#define KCB     1024
#define HW      4096       // 64*64
#define RPB     128        // rows per block
#define ZPITCH  66         // LDS pitch (even -> 8B aligned float2, bank-friendly)
#define CCHUNK  64         // codebook entries staged per pass
#define NELEM   4194304    // 16*64*64*64

// ---- codebook squared norms (K=1024, one thread per entry) ----
__global__ __launch_bounds__(256) void vq_cbnorm_kernel(const float* __restrict__ cb,
                                                        float* __restrict__ cbn) {
    int k = blockIdx.x * 256 + threadIdx.x;
    if (k < KCB) {
        const float* row = cb + (size_t)k * DIM;
        float s = 0.f;
#pragma unroll
        for (int d = 0; d < DIM; ++d) { float v = row[d]; s += v * v; }
        cbn[k] = s;
    }
}

// ---- main: WMMA f32 distance matmul + argmin + loss partial + fused output ----
// z_q_st == z_e bitwise, and the staging loop touches every z element exactly
// once across the grid, so the straight-through output store is fused here
// (non-temporal: it is never re-read) instead of a separate 32MB copy pass.
__global__ __launch_bounds__(256) void vq_argmin_kernel(const float* __restrict__ z,
                                                        const float* __restrict__ cb,
                                                        const float* __restrict__ cbn,
                                                        float* __restrict__ out,
                                                        float* __restrict__ partials) {
    __shared__ float zl[RPB * ZPITCH];      // z tile, row-major [r][d], pitch 66
    __shared__ float cl[CCHUNK * ZPITCH];   // codebook chunk [e][d], pitch 66
    __shared__ int   idxs[RPB];
    __shared__ float wsum[8];

    const int t  = threadIdx.x;
    const int L  = t & 31;       // lane in wave32
    const int w  = t >> 5;       // wave id 0..7
    const int n0 = blockIdx.x * RPB;
    const int b  = n0 >> 12;     // image index (HW=4096, RPB divides HW)
    const int hw0 = n0 & (HW - 1);
    const size_t base = (size_t)b * (DIM * HW) + hw0;
    const float* zb = z + base;
    float* outb = out + base;

    // Stage z tile with the BDHW -> (row,d) transpose folded in. Coalesced:
    // consecutive threads read consecutive hw for a fixed d (stride-4096 rows).
    // Fused straight-through output: out == z_e bitwise at identical offsets.
    for (int i = t; i < RPB * DIM; i += 256) {
        int d = i >> 7;          // /128
        int r = i & (RPB - 1);
        float val = zb[(size_t)d * HW + r];
        zl[r * ZPITCH + d] = val;
        __builtin_nontemporal_store(val, &outb[(size_t)d * HW + r]);
    }
    __syncthreads();

    // Hoist this wave's A operand (16 rows x 64 dims) into registers.
    // ISA 32-bit A 16x4 layout: lanes 0-15 hold K=0,1; lanes 16-31 hold K=2,3.
    const int arow = (w << 4) + (L & 15);
    const int koff = (L >> 4) << 1;
    v2f areg[16];
#pragma unroll
    for (int j = 0; j < 16; ++j) {
        const float* p = &zl[arow * ZPITCH + 4 * j + koff];
        v2f a; a[0] = p[0]; a[1] = p[1];
        areg[j] = a;
    }

    float minv[8];
    int   mini[8];
#pragma unroll
    for (int v = 0; v < 8; ++v) { minv[v] = 3.4e38f; mini[v] = 0; }

    for (int cb0 = 0; cb0 < KCB; cb0 += CCHUNK) {
        __syncthreads();
        // Stage CCHUNK codebook entries (coalesced row-major copy).
        for (int i = t; i < CCHUNK * DIM; i += 256) {
            int e = i >> 6;
            int d = i & (DIM - 1);
            cl[e * ZPITCH + d] = cb[(size_t)(cb0 + e) * DIM + d];
        }
        __syncthreads();

#pragma unroll
        for (int c0 = 0; c0 < CCHUNK; c0 += 16) {
            v8f acc = {};
            const int erow = c0 + (L & 15);
#pragma unroll
            for (int j = 0; j < 16; ++j) {
                const float* p = &cl[erow * ZPITCH + 4 * j + koff];
                v2f bb; bb[0] = p[0]; bb[1] = p[1];
                acc = __builtin_amdgcn_wmma_f32_16x16x4_f32(
                        false, areg[j], false, bb, (short)0, acc, false, false);
            }
            const int   cidx = cb0 + c0 + (L & 15);
            const float cn   = cbn[cidx];
#pragma unroll
            for (int v = 0; v < 8; ++v) {
                float dist = cn - 2.0f * acc[v];   // ||z||^2 constant per row: dropped
                if (dist < minv[v]) { minv[v] = dist; mini[v] = cidx; }
            }
        }
    }

    // Cross-lane argmin over the 16 lanes holding this row's columns.
    // Tie-break on the smaller index to match argmin's first-min semantics.
#pragma unroll
    for (int off = 8; off >= 1; off >>= 1) {
#pragma unroll
        for (int v = 0; v < 8; ++v) {
            float ov = __shfl_xor(minv[v], off, 32);
            int   oi = __shfl_xor(mini[v], off, 32);
            if (ov < minv[v] || (ov == minv[v] && oi < mini[v])) {
                minv[v] = ov; mini[v] = oi;
            }
        }
    }
    if ((L & 15) == 0) {
#pragma unroll
        for (int v = 0; v < 8; ++v)
            idxs[(w << 4) + v + ((L >> 4) << 3)] = mini[v];
    }
    __syncthreads();

    // Loss partial: sum (codebook[idx] - z)^2, lane handles dims 2L, 2L+1.
    float lsum = 0.f;
    for (int r = (w << 4); r < (w << 4) + 16; ++r) {
        const int idx = idxs[r];
        const float* crow = cb + (size_t)idx * DIM;
        float d0 = crow[2 * L]     - zl[r * ZPITCH + 2 * L];
        float d1 = crow[2 * L + 1] - zl[r * ZPITCH + 2 * L + 1];
        lsum += d0 * d0 + d1 * d1;
    }
#pragma unroll
    for (int off = 16; off >= 1; off >>= 1) lsum += __shfl_xor(lsum, off, 32);
    if (L == 0) wsum[w] = lsum;
    __syncthreads();
    if (t == 0) {
        float s = 0.f;
#pragma unroll
        for (int i = 0; i < 8; ++i) s += wsum[i];   // fixed order -> deterministic
        partials[blockIdx.x] = s;
    }
}

// ---- deterministic final reduction of 512 block partials -> scalar loss ----
__global__ __launch_bounds__(256) void vq_finalize_kernel(const float* __restrict__ partials,
                                                          float* __restrict__ out_scalar) {
    __shared__ float sm[256];
    int t = threadIdx.x;
    sm[t] = partials[t] + partials[t + 256];
#pragma unroll
    for (int off = 128; off >= 1; off >>= 1) {
        __syncthreads();
        if (t < off) sm[t] += sm[t + off];
    }
    __syncthreads();
    if (t == 0) out_scalar[0] = 2.0f * sm[0] / (float)NELEM;  // both MSE terms equal
}

extern "C" void kernel_launch(void* const* d_in, const int* in_sizes, int n_in,
                              void* d_out, int out_size, void* d_ws, size_t ws_size,
                              hipStream_t stream) {
    const float* z  = (const float*)d_in[0];   // [16,64,64,64]
    const float* cb = (const float*)d_in[1];   // [1024,64]
    float* out = (float*)d_out;                // 4194304 z_q_st floats + 1 loss float

    float* cbn      = (float*)d_ws;            // [1024] codebook norms
    float* partials = cbn + KCB;               // [512] per-block loss partials

    vq_cbnorm_kernel<<<4, 256, 0, stream>>>(cb, cbn);
    vq_argmin_kernel<<<NROWS / RPB, 256, 0, stream>>>(z, cb, cbn, out, partials);
    vq_finalize_kernel<<<1, 256, 0, stream>>>(partials, out + NELEM);
}